// MixtureGaussianSequenceLabeling_14156212208436
// MI455X (gfx1250) — compile-verified
//
#include <hip/hip_runtime.h>
#include <math.h>

// ---------------------------------------------------------------------------
// Mixture-Gaussian sequence labeling, CDNA5 (gfx1250).
// Latency-bound sequential scan: one workgroup (4 wave32) per (batch,dir)
// chain; all 32x32 SPD algebra done with Cholesky + f32 WMMA (16x16x4).
// ---------------------------------------------------------------------------

#define LDX 33                  // LDS row pitch for 32x32 matrices (bank-conflict free)
#define MSZ (LDX * 32)          // 1056 floats per matrix buffer
#define LOG2PI_F 1.8378770664093454f

// workspace layout (float offsets)
#define OFF_TV 0                // 2 * 64*64  transition covariances
#define OFF_S  8192             // (dir,l,b,c) scores: 2*64*16*2
#define OFF_M  12288            // means: ... * 32
#define OFF_V  143360           // covariances: ... * 1024

typedef float v2f __attribute__((ext_vector_type(2)));
typedef float v8f __attribute__((ext_vector_type(8)));

__device__ __forceinline__ void wave_sync() {
  __builtin_amdgcn_wave_barrier();
  asm volatile("s_wait_dscnt 0" ::: "memory");
  __builtin_amdgcn_wave_barrier();
}

__device__ __forceinline__ v8f wmma4(v2f a, v2f b, v8f c) {
#if defined(__has_builtin)
#if __has_builtin(__builtin_amdgcn_wmma_f32_16x16x4_f32)
  return __builtin_amdgcn_wmma_f32_16x16x4_f32(false, a, false, b, (short)0, c, false, false);
#else
  c[0] = fmaf(a.x, b.x, c[0]); return c;
#endif
#else
  c[0] = fmaf(a.x, b.x, c[0]); return c;
#endif
}

// C(32x32, pitch LDX, LDS) = op(A) * B  using v_wmma_f32_16x16x4_f32.
// A 16x4 layout: lane = 16h+m holds (M=m, K=2h+reg); B 4x16: (K=2h+reg, N=m).
__device__ __forceinline__ void mm32(const float* A, int lda, bool tA,
                                     const float* B, int ldb,
                                     float* C, int lane) {
  const int m = lane & 15, h = lane >> 4;
#pragma unroll
  for (int ti = 0; ti < 32; ti += 16) {
#pragma unroll
    for (int tj = 0; tj < 32; tj += 16) {
      v8f acc = {};
#pragma unroll
      for (int kb = 0; kb < 32; kb += 4) {
        const int k0 = kb + 2 * h;
        v2f a, bv;
        a.x = tA ? A[k0 * lda + ti + m] : A[(ti + m) * lda + k0];
        a.y = tA ? A[(k0 + 1) * lda + ti + m] : A[(ti + m) * lda + k0 + 1];
        bv.x = B[k0 * ldb + tj + m];
        bv.y = B[(k0 + 1) * ldb + tj + m];
        acc = wmma4(a, bv, acc);
      }
#pragma unroll
      for (int r = 0; r < 8; ++r)
        C[(ti + r + 8 * h) * LDX + tj + m] = acc[r];
    }
  }
}

// In-place Cholesky (lower) of SPD 32x32 in LDS; returns logdet. Wave-local.
__device__ float chol32(float* A, int lane) {
  float logdet = 0.f;
  for (int k = 0; k < 32; ++k) {
    wave_sync();
    float dk = A[k * LDX + k];
    float lkk = sqrtf(fmaxf(dk, 1e-30f));
    logdet += 2.f * logf(lkk);
    float inv = 1.f / lkk;
    float aik = (lane > k) ? A[lane * LDX + k] * inv : 0.f;
    if (lane > k) A[lane * LDX + k] = aik;
    if (lane == k) A[k * LDX + k] = lkk;
    for (int j = k + 1; j < 32; ++j) {
      float ljk = __shfl(aik, j, 32);
      if (lane >= j) A[lane * LDX + j] -= aik * ljk;
    }
  }
  wave_sync();
  return logdet;
}

// W = L^-1 (lower-triangular inverse); lane computes its own column. Wave-local.
__device__ void trinv32(const float* L, float* W, int lane) {
  for (int k = 0; k < 32; ++k) {
    if (k < lane) { W[k * LDX + lane] = 0.f; continue; }
    float s = (k == lane) ? 1.f : 0.f;
    for (int mm = lane; mm < k; ++mm) s -= L[k * LDX + mm] * W[mm * LDX + lane];
    W[k * LDX + lane] = s / L[k * LDX + k];
  }
  wave_sync();
}

__device__ __forceinline__ float mv32(const float* M, const float* x, int lane, bool tM) {
  float s = 0.f;
#pragma unroll 8
  for (int j = 0; j < 32; ++j)
    s = fmaf(tM ? M[j * LDX + lane] : M[lane * LDX + j], x[j], s);
  return s;
}

__device__ __forceinline__ float wsum32(float v) {
  for (int o = 16; o; o >>= 1) v += __shfl_xor(v, o, 32);
  return v;
}

struct WWS { float *A, *W, *P, *Q, *v0, *v1; };

// gaussian_multi: var1 dense (var1!=null) or diagonal (dvar). Returns zeta score.
__device__ float gm_unit(const float* mu0, const float* var0,
                         const float* mu1, const float* var1, const float* dvar,
                         float* outMu, float* outVar, WWS ws, int lane) {
  for (int r = 0; r < 32; ++r)
    ws.A[r * LDX + lane] = var0[r * LDX + lane] +
      (var1 ? var1[r * LDX + lane] : ((r == lane) ? dvar[lane] : 0.f));
  wave_sync();
  float logdet = chol32(ws.A, lane);
  trinv32(ws.A, ws.W, lane);
  wave_sync();
  mm32(ws.W, LDX, false, var0, LDX, ws.P, lane);               // P = W*var0
  if (var1) mm32(ws.W, LDX, false, var1, LDX, ws.Q, lane);     // Q = W*var1
  else for (int r = 0; r < 32; ++r)
    ws.Q[r * LDX + lane] = ws.W[r * LDX + lane] * dvar[lane];  // diag scale
  ws.v0[lane] = mv32(ws.W, mu0, lane, false);                  // W*mu0
  ws.v1[lane] = mv32(ws.W, mu1, lane, false);                  // W*mu1
  wave_sync();
  mm32(ws.P, LDX, true, ws.Q, LDX, outVar, lane);              // var0*inv*var1
  outMu[lane] = mv32(ws.Q, ws.v0, lane, true) + mv32(ws.P, ws.v1, lane, true);
  ws.v0[lane] = mu0[lane] - mu1[lane];
  wave_sync();
  float wd = mv32(ws.W, ws.v0, lane, false);
  float q = wsum32(wd * wd);
  return -0.5f * (32.f * LOG2PI_F + logdet + q);
}

// gaussian_multi_integral vs transition (TVt: 64x64 LDS, tmut: 64). xo/yo pick blocks.
__device__ float gmi_unit(const float* TVt, const float* tmut, int xo, int yo,
                          const float* mu1, const float* var1,
                          float* outMu, float* outVar, WWS ws, int lane) {
  for (int r = 0; r < 32; ++r)
    ws.A[r * LDX + lane] = TVt[(xo + r) * 64 + xo + lane] + var1[r * LDX + lane];
  wave_sync();
  float logdet = chol32(ws.A, lane);
  trinv32(ws.A, ws.W, lane);
  wave_sync();
  mm32(ws.W, LDX, false, TVt + xo * 64 + yo, 64, ws.P, lane);  // G = W*Sxy
  ws.v0[lane] = mu1[lane] - tmut[xo + lane];                   // diff
  wave_sync();
  mm32(ws.P, LDX, true, ws.P, LDX, ws.Q, lane);                // G^T G
  float wd = mv32(ws.W, ws.v0, lane, false);
  ws.v1[lane] = wd;
  float q = wsum32(wd * wd);
  wave_sync();
  outMu[lane] = tmut[yo + lane] + mv32(ws.P, ws.v1, lane, true);
  for (int r = 0; r < 32; ++r)
    outVar[r * LDX + lane] = TVt[(yo + r) * 64 + yo + lane] - ws.Q[r * LDX + lane];
  return -0.5f * (32.f * LOG2PI_F + logdet + q);
}

__device__ __forceinline__ void top2(const float* cs, int n, float* pidx) {
  int i0 = 0; float s0 = cs[0];
  for (int u = 1; u < n; ++u) if (cs[u] > s0) { s0 = cs[u]; i0 = u; }
  int i1 = -1; float s1 = -__builtin_inff();
  for (int u = 0; u < n; ++u) if (u != i0 && cs[u] > s1) { s1 = cs[u]; i1 = u; }
  pidx[0] = (float)i0; pidx[1] = (float)(i1 < 0 ? i0 : i1);
}

// prune top-2 of n candidates into carry AND global tape slot (dir,sl,b,c)
__device__ void prune_commit(int n, int sl, int dir, int b,
                             const float* cs, const float* cm, const float* cv,
                             float* ps, float* pm, float* pv, float* pidx,
                             float* wsS, float* wsM, float* wsV, int tid) {
  if (tid == 0) top2(cs, n, pidx);
  __syncthreads();
  for (int c = 0; c < 2; ++c) {
    int u = (int)pidx[c];
    size_t sidx = ((size_t)((dir * 64 + sl) * 16 + b)) * 2 + c;
    if (tid == 0) { ps[c] = cs[u]; wsS[sidx] = cs[u]; }
    if (tid < 32) { float mv = cm[u * 32 + tid]; pm[c * 32 + tid] = mv; wsM[sidx * 32 + tid] = mv; }
    for (int j = tid; j < 1024; j += 128) {
      int r = j >> 5, cc = j & 31;
      float vv = cv[u * MSZ + r * LDX + cc];
      pv[c * MSZ + r * LDX + cc] = vv;
      wsV[sidx * 1024 + j] = vv;
    }
  }
  __syncthreads();
}

// prune-only (final combine)
__device__ void prune_only(int n, const float* cs, const float* cm, const float* cv,
                           float* qs, float* qm, float* qv, float* pidx, int tid) {
  if (tid == 0) top2(cs, n, pidx);
  __syncthreads();
  for (int c = 0; c < 2; ++c) {
    int u = (int)pidx[c];
    if (tid == 0) qs[c] = cs[u];
    if (tid < 32) qm[c * 32 + tid] = cm[u * 32 + tid];
    for (int j = tid; j < MSZ; j += 128) qv[c * MSZ + j] = cv[u * MSZ + j];
  }
  __syncthreads();
}

// ------------------- kernel 1: TV[t] = cho[t]^T @ cho[t] (WMMA) -------------
__global__ __launch_bounds__(128) void k_transvar(const float* __restrict__ cho,
                                                  float* __restrict__ ws) {
  __shared__ float sC[4096];
  const int tid = threadIdx.x, lane = tid & 31, w = tid >> 5;
  const int m = lane & 15, h = lane >> 4;
  for (int t = 0; t < 2; ++t) {
    for (int i = tid; i < 4096; i += 128) sC[i] = cho[t * 4096 + i];
    __syncthreads();
    for (int tau = w; tau < 16; tau += 4) {
      int i0 = (tau >> 2) * 16, j0 = (tau & 3) * 16;
      v8f acc = {};
#pragma unroll
      for (int kb = 0; kb < 64; kb += 4) {
        int k0 = kb + 2 * h;
        v2f a, bv;
        a.x = sC[k0 * 64 + i0 + m];  a.y = sC[(k0 + 1) * 64 + i0 + m];  // cho^T tile
        bv.x = sC[k0 * 64 + j0 + m]; bv.y = sC[(k0 + 1) * 64 + j0 + m];
        acc = wmma4(a, bv, acc);
      }
#pragma unroll
      for (int r = 0; r < 8; ++r)
        ws[OFF_TV + t * 4096 + (i0 + r + 8 * h) * 64 + j0 + m] = acc[r];
    }
    __syncthreads();
  }
}

// ------------------- kernel 2: the 32 sequential chains ---------------------
__global__ __launch_bounds__(128) void k_scan(const int* __restrict__ sents,
    const float* __restrict__ in_mu_w, const float* __restrict__ in_cho_w,
    const float* __restrict__ trans_mu, float* __restrict__ ws) {
  extern __shared__ float sm[];
  const int tid = threadIdx.x, lane = tid & 31, w = tid >> 5;
  const int b = blockIdx.x >> 1, dir = blockIdx.x & 1;
  const int xo = dir ? 32 : 0, yo = dir ? 0 : 32;
  float* wsS = ws + OFF_S; float* wsM = ws + OFF_M; float* wsV = ws + OFF_V;

  float* p = sm;
  float* TVs = p; p += 8192;
  float* tmu = p; p += 128;
  float* ps  = p; p += 2;
  float* pm  = p; p += 64;
  float* pv  = p; p += 2 * MSZ;
  float* emu = p; p += 64;
  float* edv = p; p += 64;
  float* fsb = p; p += 4;
  float* fmb = p; p += 128;
  float* fvb = p; p += 4 * MSZ;
  float* cs  = p; p += 8;
  float* cm  = p; p += 256;
  float* cv  = p; p += 8 * MSZ;
  float* pidx = p; p += 4;
  float* ar = p + w * (4 * MSZ + 64);
  WWS wv { ar, ar + MSZ, ar + 2 * MSZ, ar + 3 * MSZ, ar + 4 * MSZ, ar + 4 * MSZ + 32 };

  for (int i = tid; i < 8192; i += 128) TVs[i] = ws[OFF_TV + i];
  tmu[tid] = trans_mu[tid];
  for (int j = tid; j < 1024; j += 128) { int r = j >> 5, c = j & 31; fvb[r * LDX + c] = (r == c) ? 1.f : 0.f; }
  if (tid < 32) fmb[tid] = 0.f;
  __syncthreads();

  // init: integrate transitions against N(0, I); prune 2 -> carry, tape slot 0
  if (w < 2) {
    float zs = gmi_unit(TVs + w * 4096, tmu + w * 64, xo, yo, fmb, fvb,
                        cm + w * 32, cv + w * MSZ, wv, lane);
    if (lane == 0) cs[w] = zs;
  }
  __syncthreads();
  prune_commit(2, 0, dir, b, cs, cm, cv, ps, pm, pv, pidx, wsS, wsM, wsV, tid);

  for (int l = 0; l < 63; ++l) {
    int pos = dir ? (63 - l) : l;
    int tok = sents[b * 64 + pos];
    if (tid < 64) {
      float mu = in_mu_w[(size_t)tok * 64 + tid];
      float sd = in_cho_w[(size_t)tok * 64 + tid];
      emu[tid] = mu; edv[tid] = sd * sd;
    }
    __syncthreads();
    { // phase 1: wave w = (c,i) emission fuse (diag emission covariance)
      int c = w >> 1, i = w & 1;
      float s = gm_unit(pm + c * 32, pv + c * MSZ, emu + i * 32, nullptr, edv + i * 32,
                        fmb + w * 32, fvb + w * MSZ, wv, lane);
      if (lane == 0) fsb[w] = s;
    }
    __syncthreads();
    // phase 2: integrate both transitions
    for (int t = 0; t < 2; ++t) {
      int u = w * 2 + t;
      float zs = gmi_unit(TVs + t * 4096, tmu + t * 64, xo, yo, fmb + w * 32, fvb + w * MSZ,
                          cm + u * 32, cv + u * MSZ, wv, lane);
      if (lane == 0) cs[u] = fsb[w] + ps[w >> 1] + zs;
    }
    __syncthreads();
    prune_commit(8, l + 1, dir, b, cs, cm, cv, ps, pm, pv, pidx, wsS, wsM, wsV, tid);
  }
}

// ------------------- kernel 3: per-(l,b) final combine + labels -------------
__global__ __launch_bounds__(128) void k_final(const int* __restrict__ sents,
    const float* __restrict__ in_mu_w, const float* __restrict__ in_cho_w,
    const float* __restrict__ out_mu, const float* __restrict__ out_cho,
    const float* __restrict__ ws, float* __restrict__ out) {
  extern __shared__ float sm[];
  const int tid = threadIdx.x, lane = tid & 31, w = tid >> 5;
  const int l = blockIdx.x >> 4, b = blockIdx.x & 15;
  const float* wsS = ws + OFF_S; const float* wsM = ws + OFF_M; const float* wsV = ws + OFF_V;

  float* p = sm;
  float* fpS = p; p += 2;  float* fpM = p; p += 64;  float* fpV = p; p += 2 * MSZ;
  float* bbS = p; p += 2;  float* bbM = p; p += 64;  float* bbV = p; p += 2 * MSZ;
  float* emu = p; p += 64; float* edv = p; p += 64;
  float* fsb = p; p += 4;  float* fmb = p; p += 128; float* fvb = p; p += 4 * MSZ;
  float* cs  = p; p += 8;  float* cm  = p; p += 256; float* cv  = p; p += 8 * MSZ;
  float* qs  = p; p += 2;  float* qm  = p; p += 64;  float* qv  = p; p += 2 * MSZ;
  float* zb  = p; p += 52; float* pidx = p; p += 4;
  float* ar = p + w * (4 * MSZ + 64);
  WWS wv { ar, ar + MSZ, ar + 2 * MSZ, ar + 3 * MSZ, ar + 4 * MSZ, ar + 4 * MSZ + 32 };

  size_t fbase = ((size_t)((0 * 64 + l) * 16 + b)) * 2;
  size_t bbase = ((size_t)((1 * 64 + (63 - l)) * 16 + b)) * 2;
  if (tid < 2) { fpS[tid] = wsS[fbase + tid]; bbS[tid] = wsS[bbase + tid]; }
  if (tid < 64) { fpM[tid] = wsM[fbase * 32 + tid]; bbM[tid] = wsM[bbase * 32 + tid]; }
  for (int j = tid; j < 2048; j += 128) {
    int c = j >> 10, r = (j >> 5) & 31, cc = j & 31;
    fpV[c * MSZ + r * LDX + cc] = wsV[(fbase + c) * 1024 + (j & 1023)];
    bbV[c * MSZ + r * LDX + cc] = wsV[(bbase + c) * 1024 + (j & 1023)];
  }
  int tok = sents[b * 64 + l];
  if (tid < 64) {
    float mu = in_mu_w[(size_t)tok * 64 + tid];
    float sd = in_cho_w[(size_t)tok * 64 + tid];
    emu[tid] = mu; edv[tid] = sd * sd;
  }
  __syncthreads();

  { // phase A: fuse forward hypothesis with emission (diag)
    int c = w >> 1, i = w & 1;
    float s = gm_unit(fpM + c * 32, fpV + c * MSZ, emu + i * 32, nullptr, edv + i * 32,
                      fmb + w * 32, fvb + w * MSZ, wv, lane);
    if (lane == 0) fsb[w] = s;
  }
  __syncthreads();
  // phase B: fuse with backward hypotheses (dense x dense)
  for (int c2 = 0; c2 < 2; ++c2) {
    int u = w * 2 + c2;
    float s = gm_unit(fmb + w * 32, fvb + w * MSZ, bbM + c2 * 32, bbV + c2 * MSZ, nullptr,
                      cm + u * 32, cv + u * MSZ, wv, lane);
    if (lane == 0) cs[u] = fsb[w] + fpS[w >> 1] + bbS[c2] + s;
  }
  __syncthreads();
  prune_only(8, cs, cm, cv, qs, qm, qv, pidx, tid);

  // phase C: 25 labels x 2 pruned comps, wave-local Cholesky each
  for (int u = w; u < 50; u += 4) {
    int pp = u & 1, n = u >> 1;
    float oc = out_cho[n * 32 + lane]; oc *= oc;
    for (int r = 0; r < 32; ++r)
      wv.A[r * LDX + lane] = qv[pp * MSZ + r * LDX + lane] + ((r == lane) ? oc : 0.f);
    wave_sync();
    float logdet = chol32(wv.A, lane);
    trinv32(wv.A, wv.W, lane);
    wave_sync();
    wv.v0[lane] = qm[pp * 32 + lane] - out_mu[n * 32 + lane];
    wave_sync();
    float wd = mv32(wv.W, wv.v0, lane, false);
    float q = wsum32(wd * wd);
    if (lane == 0)
      zb[n * 2 + pp] = -0.5f * (32.f * LOG2PI_F + logdet + q) + qs[pp];
  }
  __syncthreads();
  if (tid < 25) {
    float a = zb[tid * 2], c = zb[tid * 2 + 1];
    float mx = fmaxf(a, c);
    float r = mx + logf(expf(a - mx) + expf(c - mx));
    out[((size_t)b * 64 + l) * 25 + tid] = r;
  }
}

// ---------------------------------------------------------------------------
constexpr int SCAN_SM = (8192 + 128 + 2 + 64 + 2 * MSZ + 64 + 64 + 4 + 128 + 4 * MSZ +
                         8 + 256 + 8 * MSZ + 4 + 4 * (4 * MSZ + 64)) * 4;
constexpr int FIN_SM  = (3 * (2 + 64 + 2 * MSZ) + 128 + 4 + 128 + 4 * MSZ +
                         8 + 256 + 8 * MSZ + 52 + 4 + 4 * (4 * MSZ + 64)) * 4;

extern "C" void kernel_launch(void* const* d_in, const int* in_sizes, int n_in,
                              void* d_out, int out_size, void* d_ws, size_t ws_size,
                              hipStream_t stream) {
  (void)in_sizes; (void)n_in; (void)out_size; (void)ws_size;
  const int*   sents     = (const int*)d_in[0];
  const float* in_mu_w   = (const float*)d_in[1];
  const float* in_cho_w  = (const float*)d_in[2];
  const float* trans_mu  = (const float*)d_in[3];
  const float* trans_cho = (const float*)d_in[4];
  const float* out_mu    = (const float*)d_in[5];
  const float* out_cho   = (const float*)d_in[6];
  float* wsf  = (float*)d_ws;
  float* outp = (float*)d_out;

  (void)hipFuncSetAttribute((const void*)k_scan,
      hipFuncAttributeMaxDynamicSharedMemorySize, SCAN_SM);
  (void)hipFuncSetAttribute((const void*)k_final,
      hipFuncAttributeMaxDynamicSharedMemorySize, FIN_SM);

  k_transvar<<<1, 128, 0, stream>>>(trans_cho, wsf);
  k_scan<<<32, 128, SCAN_SM, stream>>>(sents, in_mu_w, in_cho_w, trans_mu, wsf);
  k_final<<<1024, 128, FIN_SM, stream>>>(sents, in_mu_w, in_cho_w, out_mu, out_cho, wsf, outp);
}